// StrokeRnn_14096082666012
// MI455X (gfx1250) — compile-verified
//
#include <hip/hip_runtime.h>

typedef float v2f __attribute__((ext_vector_type(2)));
typedef float v8f __attribute__((ext_vector_type(8)));

#define Hh 512
#define Bb 256
#define Ss 128

__device__ __forceinline__ float sigmoidf_(float x) {
    return 1.0f / (1.0f + expf(-x));
}

// fp32 WMMA 16x16 tile MAC: acc += A(16x4) * B(4x16), K advanced by 4 per call.
// A frag: lane l -> row (l&15), k = (l>>4)*2 + {0,1}
// B frag: lane l -> col (l&15), k = (l>>4)*2 + {0,1}
__device__ __forceinline__ v8f wmma4(v2f a, v2f b, v8f c) {
    return __builtin_amdgcn_wmma_f32_16x16x4_f32(false, a, false, b, (short)0, c,
                                                 false, false);
}

// ---------------------------------------------------------------------------
// Fused encoder LSTM step. One wave = 16 batch rows x 16 hidden cols, computes
// all 4 gates (i,f,g,o) with 4 accumulators sharing the A fragment.
// grid 64 x block 256  (512 waves = 16 M-tiles x 32 N-tiles)
// ---------------------------------------------------------------------------
__global__ void __launch_bounds__(256)
enc_step(const float* __restrict__ x, int t,
         const float* __restrict__ Wih,  // (4H,2)
         const float* __restrict__ bih, const float* __restrict__ bhh,
         const float* __restrict__ Whh,  // (4H,H)
         const float* __restrict__ h_in, const float* __restrict__ c_in,
         float* __restrict__ h_out, float* __restrict__ c_out,
         const int* __restrict__ seq_len, float* __restrict__ hn) {
    int wave = blockIdx.x * 8 + (threadIdx.x >> 5);
    int lane = threadIdx.x & 31;
    int mt = wave >> 5;   // 0..15
    int nt = wave & 31;   // 0..31
    int mrow  = (mt << 4) + (lane & 15);
    int khalf = (lane >> 4) << 1;
    int ncol  = (nt << 4) + (lane & 15);

    const float* Arow = h_in + mrow * Hh;
    const float* W0 = Whh + (size_t)(0 * Hh + ncol) * Hh;
    const float* W1 = Whh + (size_t)(1 * Hh + ncol) * Hh;
    const float* W2 = Whh + (size_t)(2 * Hh + ncol) * Hh;
    const float* W3 = Whh + (size_t)(3 * Hh + ncol) * Hh;

    v8f ai = {}, af = {}, ag = {}, ao = {};
    for (int k = 0; k < Hh; k += 4) {
        int kk = k + khalf;
        v2f a  = *(const v2f*)(Arow + kk);
        v2f b0 = *(const v2f*)(W0 + kk);
        v2f b1 = *(const v2f*)(W1 + kk);
        v2f b2 = *(const v2f*)(W2 + kk);
        v2f b3 = *(const v2f*)(W3 + kk);
        ai = wmma4(a, b0, ai);
        af = wmma4(a, b1, af);
        ag = wmma4(a, b2, ag);
        ao = wmma4(a, b3, ao);
    }

    const float* xt = x + (size_t)t * Bb * 2;
    int j0 = ncol, j1 = Hh + ncol, j2 = 2 * Hh + ncol, j3 = 3 * Hh + ncol;
    float bi = bih[j0] + bhh[j0], bf = bih[j1] + bhh[j1];
    float bg = bih[j2] + bhh[j2], bo = bih[j3] + bhh[j3];
    float wi0 = Wih[j0 * 2], wi1 = Wih[j0 * 2 + 1];
    float wf0 = Wih[j1 * 2], wf1 = Wih[j1 * 2 + 1];
    float wg0 = Wih[j2 * 2], wg1 = Wih[j2 * 2 + 1];
    float wo0 = Wih[j3 * 2], wo1 = Wih[j3 * 2 + 1];

#pragma unroll
    for (int e = 0; e < 8; ++e) {
        int row = (mt << 4) + e + ((lane >> 4) << 3);
        float x0 = xt[row * 2], x1 = xt[row * 2 + 1];
        float gi = ai[e] + bi + x0 * wi0 + x1 * wi1;
        float gf = af[e] + bf + x0 * wf0 + x1 * wf1;
        float gg = ag[e] + bg + x0 * wg0 + x1 * wg1;
        float go = ao[e] + bo + x0 * wo0 + x1 * wo1;
        float iv = sigmoidf_(gi), fv = sigmoidf_(gf);
        float gv = tanhf(gg),    ov = sigmoidf_(go);
        float cn = fv * c_in[row * Hh + ncol] + iv * gv;
        float hv = ov * tanhf(cn);
        c_out[row * Hh + ncol] = cn;
        h_out[row * Hh + ncol] = hv;
        if (t == seq_len[row] - 1) hn[row * Hh + ncol] = hv;
    }
}

// ---------------------------------------------------------------------------
// Fused decoder LSTM step. Pc = code @ Wih_code^T + bih + bhh (precomputed).
// Per-step input contribution is only prev(B,2) @ Wih[:, :2]^T.
// ---------------------------------------------------------------------------
__global__ void __launch_bounds__(256)
dec_step(const float* __restrict__ Pc,   // (B,4H)
         const float* __restrict__ Wih,  // dec_Wih (4H,514), cols 0..1 used here
         const float* __restrict__ Whh,  // (4H,H)
         const float* __restrict__ prev, // (B,2)
         const float* __restrict__ h_in, const float* __restrict__ c_in,
         float* __restrict__ h_out, float* __restrict__ c_out) {
    int wave = blockIdx.x * 8 + (threadIdx.x >> 5);
    int lane = threadIdx.x & 31;
    int mt = wave >> 5;
    int nt = wave & 31;
    int mrow  = (mt << 4) + (lane & 15);
    int khalf = (lane >> 4) << 1;
    int ncol  = (nt << 4) + (lane & 15);

    const float* Arow = h_in + mrow * Hh;
    const float* W0 = Whh + (size_t)(0 * Hh + ncol) * Hh;
    const float* W1 = Whh + (size_t)(1 * Hh + ncol) * Hh;
    const float* W2 = Whh + (size_t)(2 * Hh + ncol) * Hh;
    const float* W3 = Whh + (size_t)(3 * Hh + ncol) * Hh;

    v8f ai = {}, af = {}, ag = {}, ao = {};
    for (int k = 0; k < Hh; k += 4) {
        int kk = k + khalf;
        v2f a  = *(const v2f*)(Arow + kk);
        v2f b0 = *(const v2f*)(W0 + kk);
        v2f b1 = *(const v2f*)(W1 + kk);
        v2f b2 = *(const v2f*)(W2 + kk);
        v2f b3 = *(const v2f*)(W3 + kk);
        ai = wmma4(a, b0, ai);
        af = wmma4(a, b1, af);
        ag = wmma4(a, b2, ag);
        ao = wmma4(a, b3, ao);
    }

    int j0 = ncol, j1 = Hh + ncol, j2 = 2 * Hh + ncol, j3 = 3 * Hh + ncol;
    float wi0 = Wih[(size_t)j0 * 514], wi1 = Wih[(size_t)j0 * 514 + 1];
    float wf0 = Wih[(size_t)j1 * 514], wf1 = Wih[(size_t)j1 * 514 + 1];
    float wg0 = Wih[(size_t)j2 * 514], wg1 = Wih[(size_t)j2 * 514 + 1];
    float wo0 = Wih[(size_t)j3 * 514], wo1 = Wih[(size_t)j3 * 514 + 1];

#pragma unroll
    for (int e = 0; e < 8; ++e) {
        int row = (mt << 4) + e + ((lane >> 4) << 3);
        float p0 = prev[row * 2], p1 = prev[row * 2 + 1];
        const float* pcr = Pc + (size_t)row * (4 * Hh);
        float gi = ai[e] + pcr[j0] + p0 * wi0 + p1 * wi1;
        float gf = af[e] + pcr[j1] + p0 * wf0 + p1 * wf1;
        float gg = ag[e] + pcr[j2] + p0 * wg0 + p1 * wg1;
        float go = ao[e] + pcr[j3] + p0 * wo0 + p1 * wo1;
        float iv = sigmoidf_(gi), fv = sigmoidf_(gf);
        float gv = tanhf(gg),    ov = sigmoidf_(go);
        float cn = fv * c_in[row * Hh + ncol] + iv * gv;
        float hv = ov * tanhf(cn);
        c_out[row * Hh + ncol] = cn;
        h_out[row * Hh + ncol] = hv;
    }
}

// ---------------------------------------------------------------------------
// Generic WMMA GEMM: C(256,N) = act( A(256,512) @ W(N,:)^T + b1 + b2 )
// W element (n,k) at W[n*ldw + koff + k].  act: 0=identity, 1=tanh
// grid = (16 * N/16)/8 blocks of 256
// ---------------------------------------------------------------------------
__global__ void __launch_bounds__(256)
gemm_nt(const float* __restrict__ A, const float* __restrict__ W, int ldw,
        int koff, const float* __restrict__ b1, const float* __restrict__ b2,
        float* __restrict__ C, int N, int act) {
    int wave = blockIdx.x * 8 + (threadIdx.x >> 5);
    int lane = threadIdx.x & 31;
    int ntiles = N >> 4;
    int mt = wave / ntiles;
    int nt = wave % ntiles;
    int mrow  = (mt << 4) + (lane & 15);
    int khalf = (lane >> 4) << 1;
    int ncol  = (nt << 4) + (lane & 15);

    const float* Arow = A + mrow * Hh;
    const float* Wrow = W + (size_t)ncol * ldw + koff;

    v8f acc = {};
    for (int k = 0; k < Hh; k += 4) {
        int kk = k + khalf;
        v2f a = *(const v2f*)(Arow + kk);
        v2f b = *(const v2f*)(Wrow + kk);
        acc = wmma4(a, b, acc);
    }

    float bias = (b1 ? b1[ncol] : 0.0f) + (b2 ? b2[ncol] : 0.0f);
#pragma unroll
    for (int e = 0; e < 8; ++e) {
        int row = (mt << 4) + e + ((lane >> 4) << 3);
        float v = acc[e] + bias;
        if (act == 1) v = tanhf(v);
        C[(size_t)row * N + ncol] = v;
    }
}

// code = mean + exp(logstd/2) * epsilon
__global__ void make_code(const float* __restrict__ mean,
                          const float* __restrict__ logstd,
                          const float* __restrict__ eps,
                          float* __restrict__ code) {
    int i = blockIdx.x * blockDim.x + threadIdx.x;
    if (i < Bb * Hh) code[i] = mean[i] + expf(0.5f * logstd[i]) * eps[i];
}

// o(B,2) = h(B,512) @ out_W(2,512)^T + out_b ; also feeds prev for next step
__global__ void out_proj(const float* __restrict__ h,
                         const float* __restrict__ outW,
                         const float* __restrict__ outb,
                         float* __restrict__ dec_out_t,
                         float* __restrict__ prev) {
    int idx = blockIdx.x * blockDim.x + threadIdx.x;
    if (idx >= Bb * 2) return;
    int b = idx >> 1, oc = idx & 1;
    const float* hr = h + (size_t)b * Hh;
    const float* wr = outW + (size_t)oc * Hh;
    float s = outb[oc];
#pragma unroll 4
    for (int k = 0; k < Hh; ++k) s += hr[k] * wr[k];
    dec_out_t[b * 2 + oc] = s;
    prev[b * 2 + oc] = s;
}

__global__ void zero_f32(float* __restrict__ p, int n) {
    int i = blockIdx.x * blockDim.x + threadIdx.x;
    if (i < n) p[i] = 0.0f;
}

extern "C" void kernel_launch(void* const* d_in, const int* in_sizes, int n_in,
                              void* d_out, int out_size, void* d_ws, size_t ws_size,
                              hipStream_t stream) {
    const float* x        = (const float*)d_in[0];
    const int*   seq_len  = (const int*)d_in[1];
    const float* eps      = (const float*)d_in[2];
    const float* enc_Wih  = (const float*)d_in[3];
    const float* enc_Whh  = (const float*)d_in[4];
    const float* enc_bih  = (const float*)d_in[5];
    const float* enc_bhh  = (const float*)d_in[6];
    const float* mean_W   = (const float*)d_in[7];
    const float* mean_b   = (const float*)d_in[8];
    const float* logstd_W = (const float*)d_in[9];
    const float* logstd_b = (const float*)d_in[10];
    const float* decH_W   = (const float*)d_in[11];
    const float* decH_b   = (const float*)d_in[12];
    const float* decC_W   = (const float*)d_in[13];
    const float* decC_b   = (const float*)d_in[14];
    const float* dec_Wih  = (const float*)d_in[15];
    const float* dec_Whh  = (const float*)d_in[16];
    const float* dec_bih  = (const float*)d_in[17];
    const float* dec_bhh  = (const float*)d_in[18];
    const float* out_W    = (const float*)d_in[19];
    const float* out_b    = (const float*)d_in[20];

    float* out = (float*)d_out;
    float* ws  = (float*)d_ws;

    const int BH = Bb * Hh;  // 131072
    float* h0   = ws;
    float* c0   = ws + BH;
    float* h1   = ws + 2 * BH;
    float* c1   = ws + 3 * BH;
    float* hn   = ws + 4 * BH;
    float* code = ws + 5 * BH;
    float* Pc   = ws + 6 * BH;       // (B, 4H) = 4*BH
    float* prev = ws + 10 * BH;      // (B, 2)

    float* mean_o   = out + Ss * Bb * 2;     // d_out: dec_out | mean | logstd
    float* logstd_o = mean_o + BH;

    // init h0,c0 (adjacent) and prev to zero
    zero_f32<<<(2 * BH + 255) / 256, 256, 0, stream>>>(h0, 2 * BH);
    zero_f32<<<2, 256, 0, stream>>>(prev, Bb * 2);

    // ---- Encoder: 128 fused LSTM steps, hn captured at seq_len-1 ----
    for (int t = 0; t < Ss; ++t) {
        const float* hi = (t & 1) ? h1 : h0;
        const float* ci = (t & 1) ? c1 : c0;
        float* ho = (t & 1) ? h0 : h1;
        float* co = (t & 1) ? c0 : c1;
        enc_step<<<64, 256, 0, stream>>>(x, t, enc_Wih, enc_bih, enc_bhh,
                                         enc_Whh, hi, ci, ho, co, seq_len, hn);
    }

    // ---- Latent: mean / logstd -> code ----
    gemm_nt<<<64, 256, 0, stream>>>(hn, mean_W, Hh, 0, mean_b, nullptr,
                                    mean_o, Hh, 0);
    gemm_nt<<<64, 256, 0, stream>>>(hn, logstd_W, Hh, 0, logstd_b, nullptr,
                                    logstd_o, Hh, 0);
    make_code<<<(BH + 255) / 256, 256, 0, stream>>>(mean_o, logstd_o, eps, code);

    // ---- Decoder init: hd -> h0, cd -> c0 (tanh epilogue) ----
    gemm_nt<<<64, 256, 0, stream>>>(code, decH_W, Hh, 0, decH_b, nullptr,
                                    h0, Hh, 1);
    gemm_nt<<<64, 256, 0, stream>>>(code, decC_W, Hh, 0, decC_b, nullptr,
                                    c0, Hh, 1);

    // ---- Precompute constant input contribution:
    // Pc = code @ dec_Wih[:,2:]^T + dec_bih + dec_bhh  (B, 4H) ----
    gemm_nt<<<256, 256, 0, stream>>>(code, dec_Wih, 514, 2, dec_bih, dec_bhh,
                                     Pc, 4 * Hh, 0);

    // ---- Decoder: 128 fused LSTM steps + O=2 output projection ----
    for (int t = 0; t < Ss; ++t) {
        const float* hi = (t & 1) ? h1 : h0;
        const float* ci = (t & 1) ? c1 : c0;
        float* ho = (t & 1) ? h0 : h1;
        float* co = (t & 1) ? c0 : c1;
        dec_step<<<64, 256, 0, stream>>>(Pc, dec_Wih, dec_Whh, prev,
                                         hi, ci, ho, co);
        out_proj<<<2, 256, 0, stream>>>(ho, out_W, out_b, out + (size_t)t * Bb * 2,
                                        prev);
    }
}